// NodeGCN_29394756174095
// MI455X (gfx1250) — compile-verified
//
#include <hip/hip_runtime.h>
#include <math.h>

#define GCN_N 50000
#define GCN_E 800000
#define GCN_DIN 96
#define GCN_DHID 96
#define GCN_DOUT 32

typedef float v2f __attribute__((ext_vector_type(2)));
typedef float v8f __attribute__((ext_vector_type(8)));

// ---------------------------------------------------------------- degree prep
__global__ void k_deg_init(float* __restrict__ deg, int n) {
    int i = blockIdx.x * blockDim.x + threadIdx.x;
    if (i < n) deg[i] = 1.0f;   // self-loop
}

__global__ void k_edges_prep(const long long* __restrict__ ei,
                             int* __restrict__ src32, int* __restrict__ dst32,
                             float* __restrict__ deg, int e) {
    int i = blockIdx.x * blockDim.x + threadIdx.x;
    if (i < e) {
        int s = (int)ei[i];
        int d = (int)ei[(size_t)e + i];
        src32[i] = s;
        dst32[i] = d;
        atomicAdd(&deg[d], 1.0f);   // exact: integer-valued float sums
    }
}

__global__ void k_dinv(float* __restrict__ deg, int n) {
    int i = blockIdx.x * blockDim.x + threadIdx.x;
    if (i < n) {
        float dg = deg[i];
        deg[i] = (dg > 0.0f) ? rsqrtf(dg) : 0.0f;
    }
}

__global__ void k_norm(const int* __restrict__ src, const int* __restrict__ dst,
                       const float* __restrict__ dinv, float* __restrict__ norm, int e) {
    int i = blockIdx.x * blockDim.x + threadIdx.x;
    if (i < e) norm[i] = dinv[src[i]] * dinv[dst[i]];
}

// ---------------------------------------------------------------- WMMA GEMM
// C[M x DOUT] = A[M x KDIM] @ W[KDIM x DOUT], fp32 exact via V_WMMA_F32_16X16X4_F32.
// W staged in LDS in PAIR-INTERLEAVED layout:
//   Wlds[(p*DOUT + c)*2 + j] = W[2p + j][c]        (p = K-pair index)
// so each lane's B fragment {W[kb][c], W[kb+1][c]} is one contiguous 8B chunk:
// a single ds_load_b64 into an even-aligned VGPR pair feeding WMMA directly.
// One wave = one 16-row tile x ALL column tiles -> A frag (one global_load_b64
// per k-step) reused DOUT/16 times.
// Fragment layouts (ISA 7.12.2, 32-bit):
//   A 16x4: lane half h holds K = k0+2h (+0,+1 in the two VGPRs), row = lane&15
//   B 4x16: VGPR v = row K = k0+2h+v, col = lane&15
//   C 16x16: VGPR r -> row r+8h, col = lane&15
template <int KDIM, int DOUT, bool RELU_IN>
__global__ __launch_bounds__(256) void k_gemm_wmma(const float* __restrict__ A,
                                                   const float* __restrict__ W,
                                                   float* __restrict__ C,
                                                   int nrowtiles) {
    constexpr int NT    = DOUT / 16;         // column tiles per wave (6 or 2)
    constexpr int NPAIR = KDIM / 2;          // K pairs
    __shared__ float Wlds[KDIM * DOUT];

    // cooperative W -> LDS with pair interleave
    for (int i = threadIdx.x; i < NPAIR * DOUT; i += 256) {
        int p = i / DOUT, c = i - p * DOUT;
        float2 w;
        w.x = W[(2 * p) * DOUT + c];
        w.y = W[(2 * p + 1) * DOUT + c];
        ((float2*)Wlds)[i] = w;
    }
    __syncthreads();

    int wave = (blockIdx.x * 256 + threadIdx.x) >> 5;
    int lane = threadIdx.x & 31;
    if (wave >= nrowtiles) return;           // wave-uniform: EXEC all-ones below

    int half = lane >> 4;                    // 0 or 1
    int l16  = lane & 15;

    const float* ap = A + (unsigned)(wave * 16 + l16) * KDIM + 2u * (unsigned)half;
    // lane's B base: pair p = k0/2 + half, col = t*16 + l16
    const float* bp = Wlds + ((unsigned)half * DOUT + l16) * 2u;

    v8f acc[NT];
#pragma unroll
    for (int t = 0; t < NT; ++t) acc[t] = (v8f)0.0f;

#pragma unroll
    for (int k0 = 0; k0 < KDIM; k0 += 4) {
        float2 av = *(const float2*)(ap + k0);      // global_load_b64, 8B aligned
        if (RELU_IN) {                              // fuse prev layer's ReLU
            av.x = fmaxf(av.x, 0.0f);
            av.y = fmaxf(av.y, 0.0f);
        }
        v2f a; a.x = av.x; a.y = av.y;
#pragma unroll
        for (int t = 0; t < NT; ++t) {
            // one ds_load_b64, immediate offset = (k0/2)*DOUT*8 + t*128 bytes
            v2f b = *(const v2f*)(bp + (k0 / 2) * DOUT * 2 + t * 32);
            acc[t] = __builtin_amdgcn_wmma_f32_16x16x4_f32(
                false, a, false, b, (short)0, acc[t], false, false);
        }
    }

    float* cp = C + (unsigned)(wave * 16 + 8 * half) * DOUT + l16;
#pragma unroll
    for (int t = 0; t < NT; ++t)
#pragma unroll
        for (int r = 0; r < 8; ++r)
            cp[r * DOUT + t * 16] = acc[t][r];
}

// ---------------------------------------------------------------- aggregation
// out[i,f] = bias[f] + dinv[i]^2 * h[i,f]   (self-loop term + bias)
template <int D>
__global__ void k_agg_init(const float* __restrict__ h, const float* __restrict__ dinv,
                           const float* __restrict__ bias, float* __restrict__ out,
                           int n) {
    int t = blockIdx.x * blockDim.x + threadIdx.x;
    if (t < n * D) {
        int row = t / D, col = t - row * D;
        float di = dinv[row];
        out[t] = bias[col] + di * di * h[t];
    }
}

// out[dst] += norm[e] * h[src]  — float4 gather (L2 resident) + 4 atomic adds
template <int D>
__global__ void k_agg_edges(const int* __restrict__ src, const int* __restrict__ dst,
                            const float* __restrict__ norm, const float* __restrict__ h,
                            float* __restrict__ out, int e) {
    constexpr int CH = D / 4;
    int t = blockIdx.x * blockDim.x + threadIdx.x;
    if (t >= e * CH) return;
    int ed = t / CH;
    int c  = (t - ed * CH) << 2;
    int s  = src[ed], dd = dst[ed];
    float nm = norm[ed];
    const float4 hv = *(const float4*)(h + (unsigned)s * D + c);
    float* op = out + (unsigned)dd * D + c;
    atomicAdd(op + 0, nm * hv.x);
    atomicAdd(op + 1, nm * hv.y);
    atomicAdd(op + 2, nm * hv.z);
    atomicAdd(op + 3, nm * hv.w);
}

// ---------------------------------------------------------------- launch
static inline size_t align256(size_t x) { return (x + 255) & ~(size_t)255; }

extern "C" void kernel_launch(void* const* d_in, const int* in_sizes, int n_in,
                              void* d_out, int out_size, void* d_ws, size_t ws_size,
                              hipStream_t stream) {
    (void)in_sizes; (void)n_in; (void)out_size; (void)ws_size;
    const float*     x  = (const float*)d_in[0];
    const long long* ei = (const long long*)d_in[1];   // int64 [2, E]
    const float*     W1 = (const float*)d_in[2];
    const float*     b1 = (const float*)d_in[3];
    const float*     W2 = (const float*)d_in[4];
    const float*     b2 = (const float*)d_in[5];
    const float*     W3 = (const float*)d_in[6];
    const float*     b3 = (const float*)d_in[7];
    float* out = (float*)d_out;

    // workspace carve-out
    char* ws = (char*)d_ws;
    size_t off = 0;
    float* dinv  = (float*)(ws + off); off += align256((size_t)GCN_N * 4);
    int*   src32 = (int*)  (ws + off); off += align256((size_t)GCN_E * 4);
    int*   dst32 = (int*)  (ws + off); off += align256((size_t)GCN_E * 4);
    float* norm  = (float*)(ws + off); off += align256((size_t)GCN_E * 4);
    float* buf0  = (float*)(ws + off); off += align256((size_t)GCN_N * GCN_DHID * 4); // h (post-GEMM)
    float* buf1  = (float*)(ws + off); off += align256((size_t)GCN_N * GCN_DHID * 4); // aggregated
    float* buf2  = (float*)(ws + off); off += align256((size_t)GCN_N * GCN_DOUT * 4); // h3

    const int T = 256;
    auto blk = [](long long n, int t) { return (int)((n + t - 1) / t); };

    // --- normalization precompute (shared by all 3 layers) ---
    k_deg_init<<<blk(GCN_N, T), T, 0, stream>>>(dinv, GCN_N);
    k_edges_prep<<<blk(GCN_E, T), T, 0, stream>>>(ei, src32, dst32, dinv, GCN_E);
    k_dinv<<<blk(GCN_N, T), T, 0, stream>>>(dinv, GCN_N);
    k_norm<<<blk(GCN_E, T), T, 0, stream>>>(src32, dst32, dinv, norm, GCN_E);

    const int rowtiles = GCN_N / 16;                  // 3125 waves per GEMM
    const int gemm_blocks = blk((long long)rowtiles * 32, T);

    // --- layer 1: h1 = x@W1 ; agg -> buf1 (ReLU deferred into next GEMM) ---
    k_gemm_wmma<GCN_DIN, GCN_DHID, false><<<gemm_blocks, T, 0, stream>>>(
        x, W1, buf0, rowtiles);
    k_agg_init<GCN_DHID><<<blk((long long)GCN_N * GCN_DHID, T), T, 0, stream>>>(
        buf0, dinv, b1, buf1, GCN_N);
    k_agg_edges<GCN_DHID><<<blk((long long)GCN_E * (GCN_DHID / 4), T), T, 0, stream>>>(
        src32, dst32, norm, buf0, buf1, GCN_E);

    // --- layer 2: h2 = relu(buf1)@W2 ; agg -> buf1 ---
    k_gemm_wmma<GCN_DHID, GCN_DHID, true><<<gemm_blocks, T, 0, stream>>>(
        buf1, W2, buf0, rowtiles);
    k_agg_init<GCN_DHID><<<blk((long long)GCN_N * GCN_DHID, T), T, 0, stream>>>(
        buf0, dinv, b2, buf1, GCN_N);
    k_agg_edges<GCN_DHID><<<blk((long long)GCN_E * (GCN_DHID / 4), T), T, 0, stream>>>(
        src32, dst32, norm, buf0, buf1, GCN_E);

    // --- layer 3: h3 = relu(buf1)@W3 ; agg -> d_out ---
    k_gemm_wmma<GCN_DHID, GCN_DOUT, true><<<gemm_blocks, T, 0, stream>>>(
        buf1, W3, buf2, rowtiles);
    k_agg_init<GCN_DOUT><<<blk((long long)GCN_N * GCN_DOUT, T), T, 0, stream>>>(
        buf2, dinv, b3, out, GCN_N);
    k_agg_edges<GCN_DOUT><<<blk((long long)GCN_E * (GCN_DOUT / 4), T), T, 0, stream>>>(
        src32, dst32, norm, buf2, out, GCN_E);
}